// DecoderRNN_18193481466337
// MI455X (gfx1250) — compile-verified
//
#include <hip/hip_runtime.h>
#include <hip/hip_bf16.h>
#include <math.h>

// ---------------------------------------------------------------------------
// DecoderRNN single step on MI455X (gfx1250, wave32).
//
// Roofline: all matrix ops are batch-1 GEMVs; weight traffic = 546 MB f32 ->
// HBM-bound (~23 us at 23.3 TB/s). Weights are streamed with non-temporal
// loads; V_WMMA_F32_16X16X4_F32 performs the 16-row x K=4 contraction per
// instruction (cross-lane K-sum in the matrix pipe, no shuffle tree).
//
// Concurrency: ~16 MB of loads must be in flight to cover HBM latency
// (23.3 TB/s x ~700 ns). Each wave holds ~4.3 KB outstanding, so we need
// ~4000 waves. Both GEMVs are K-split to raise wave count:
//   gates : 256 row-tiles x 8 K-segs = 2048 waves  (134 MB)
//   logits: 1571 row-tiles x 4 K-segs = 6284 waves (412 MB, dominant)
// Partials are summed in fixed order (deterministic, no FP atomics).
// ---------------------------------------------------------------------------

#define VOCAB   50257
#define VPAD    50264            // VOCAB padded to a multiple of 8
#define HIDDEN  2048
#define GATES   (4 * HIDDEN)
#define KSPLIT_G 8               // K-split for gates GEMV
#define KSPLIT_V 4               // K-split for logits GEMV

typedef __attribute__((ext_vector_type(2))) float v2f;
typedef __attribute__((ext_vector_type(8))) float v8f;

// Per-wave: accumulate 32 weight rows (two 16-row tiles) against vec[k0..k1)
// into c0/c1 with V_WMMA_F32_16X16X4_F32.
// A 16x4 f32 layout (ISA 7.12.2): lane m (0..15) holds (M=m, K=0..1);
// lane m+16 holds (M=m, K=2..3). B 4x16: lanes 0..15 hold K=0..1, lanes
// 16..31 hold K=2..3 -> identical per-lane float2 addressing; the activation
// chunk is replicated across all 16 B columns so every D column carries the
// same 16 dot products.
__device__ __forceinline__ void gemv_tile_accum(
    const float* __restrict__ W, const float* __restrict__ vec,
    int rowA, int rowB, int k0, int k1, int lhalf, v8f& c0, v8f& c1)
{
  const float* a0p = W + (size_t)rowA * HIDDEN;
  const float* a1p = W + (size_t)rowB * HIDDEN;
  for (int k = k0; k < k1; k += 4) {
    const int ka = k + lhalf * 2;                     // 8B aligned (k%4==0)
    v2f a0 = __builtin_nontemporal_load((const v2f*)(a0p + ka));  // stream
    v2f a1 = __builtin_nontemporal_load((const v2f*)(a1p + ka));  // stream
    v2f b  = *(const v2f*)(vec + ka);                 // hot, cacheable
    c0 = __builtin_amdgcn_wmma_f32_16x16x4_f32(false, a0, false, b,
                                               (short)0, c0, false, false);
    c1 = __builtin_amdgcn_wmma_f32_16x16x4_f32(false, a1, false, b,
                                               (short)0, c1, false, false);
  }
}

// ---- 1) x = relu(emb[token]) -----------------------------------------------
__global__ void embed_relu_kernel(const int* __restrict__ token,
                                  const float* __restrict__ emb,
                                  float* __restrict__ x)
{
  const int k = blockIdx.x * blockDim.x + threadIdx.x;
  if (k < HIDDEN) {
    const float e = emb[(size_t)token[0] * HIDDEN + k];
    x[k] = fmaxf(e, 0.0f);
  }
}

// ---- 2) gates partials: gpart[s][j] = dot over K-seg s of Wih[j]*x+Whh[j]*h -
__global__ void gates_gemv_kernel(const float* __restrict__ Wih,
                                  const float* __restrict__ Whh,
                                  const float* __restrict__ x,
                                  const float* __restrict__ h,
                                  float* __restrict__ gpart)
{
  const int wave = (blockIdx.x * blockDim.x + threadIdx.x) >> 5;
  const int lane = threadIdx.x & 31;
  const int tile = wave >> 3;            // /KSPLIT_G, uniform per wave
  const int kseg = wave & (KSPLIT_G - 1);
  if (tile >= GATES / 32) return;        // wave-uniform guard
  const int row0  = tile * 32;
  const int k0    = kseg * (HIDDEN / KSPLIT_G);
  const int k1    = k0 + (HIDDEN / KSPLIT_G);
  const int lhalf = lane >> 4;
  const int lrow  = lane & 15;

  v8f c0 = {};
  v8f c1 = {};
  gemv_tile_accum(Wih, x, row0 + lrow, row0 + 16 + lrow, k0, k1, lhalf, c0, c1);
  gemv_tile_accum(Whh, h, row0 + lrow, row0 + 16 + lrow, k0, k1, lhalf, c0, c1);

  // D layout: VGPR j = row j (lanes 0-15) / row j+8 (lanes 16-31); all
  // columns identical since B was replicated -> lanes 0 and 16 hold results.
  float* out = gpart + (size_t)kseg * GATES + row0;
  if (lane == 0) {
#pragma unroll
    for (int j = 0; j < 8; ++j) { out[j]     = c0[j]; out[16 + j] = c1[j]; }
  } else if (lane == 16) {
#pragma unroll
    for (int j = 0; j < 8; ++j) { out[8 + j] = c0[j]; out[24 + j] = c1[j]; }
  }
}

// ---- 3) LSTM pointwise: h_new, c_new ---------------------------------------
__global__ void lstm_cell_kernel(const float* __restrict__ gpart,
                                 const float* __restrict__ b_ih,
                                 const float* __restrict__ b_hh,
                                 const float* __restrict__ c_old,
                                 float* __restrict__ out,      // d_out base
                                 float* __restrict__ h_copy)   // aligned ws copy
{
  const int k = blockIdx.x * blockDim.x + threadIdx.x;
  if (k >= HIDDEN) return;
  float gi = b_ih[k]              + b_hh[k];
  float gf = b_ih[HIDDEN + k]     + b_hh[HIDDEN + k];
  float gg = b_ih[2 * HIDDEN + k] + b_hh[2 * HIDDEN + k];
  float go = b_ih[3 * HIDDEN + k] + b_hh[3 * HIDDEN + k];
#pragma unroll
  for (int s = 0; s < KSPLIT_G; ++s) {
    const float* gp = gpart + (size_t)s * GATES;
    gi += gp[k];
    gf += gp[HIDDEN + k];
    gg += gp[2 * HIDDEN + k];
    go += gp[3 * HIDDEN + k];
  }
  const float i = 1.0f / (1.0f + expf(-gi));
  const float f = 1.0f / (1.0f + expf(-gf));
  const float g = tanhf(gg);
  const float o = 1.0f / (1.0f + expf(-go));
  const float cn = f * c_old[k] + i * g;
  const float hn = o * tanhf(cn);
  out[VOCAB + k]          = hn;   // h_new
  out[VOCAB + HIDDEN + k] = cn;   // c_new
  h_copy[k] = hn;                 // 8B-aligned copy for the logits GEMV
}

// ---- 4) logits partials: lpart[s][v] = dot over K-seg s of W_out[v]*h_new ---
__global__ void logits_gemv_kernel(const float* __restrict__ Wout,
                                   const float* __restrict__ hn,
                                   float* __restrict__ lpart)
{
  const int wave = (blockIdx.x * blockDim.x + threadIdx.x) >> 5;
  const int lane = threadIdx.x & 31;
  const int NT = (VOCAB + 31) / 32;            // 1571 row tiles
  const int tile = wave >> 2;                  // /KSPLIT_V, uniform per wave
  const int kseg = wave & (KSPLIT_V - 1);
  if (tile >= NT) return;                      // wave-uniform
  const int row0  = tile * 32;
  const int k0    = kseg * (HIDDEN / KSPLIT_V);
  const int k1    = k0 + (HIDDEN / KSPLIT_V);
  const int lhalf = lane >> 4;
  const int lrow  = lane & 15;
  const int rA = min(row0 + lrow,      VOCAB - 1);   // clamp tail-tile loads
  const int rB = min(row0 + 16 + lrow, VOCAB - 1);

  v8f c0 = {};
  v8f c1 = {};
  gemv_tile_accum(Wout, hn, rA, rB, k0, k1, lhalf, c0, c1);

  float* out = lpart + (size_t)kseg * VPAD;
  if (lane == 0) {
#pragma unroll
    for (int j = 0; j < 8; ++j) {
      int r = row0 + j;      if (r < VOCAB) out[r] = c0[j];
      r = row0 + 16 + j;     if (r < VOCAB) out[r] = c1[j];
    }
  } else if (lane == 16) {
#pragma unroll
    for (int j = 0; j < 8; ++j) {
      int r = row0 + 8 + j;  if (r < VOCAB) out[r] = c0[j];
      r = row0 + 24 + j;     if (r < VOCAB) out[r] = c1[j];
    }
  }
}

// ---- 5) combine logits partials + bias -------------------------------------
__global__ void logits_combine_kernel(const float* __restrict__ lpart,
                                      const float* __restrict__ bout,
                                      float* __restrict__ logits)
{
  const int v = blockIdx.x * blockDim.x + threadIdx.x;
  if (v < VOCAB) {
    float s = bout[v];
#pragma unroll
    for (int p = 0; p < KSPLIT_V; ++p) s += lpart[(size_t)p * VPAD + v];
    logits[v] = s;
  }
}

// ---- 6) single-block max / log-sum-exp over VOCAB --------------------------
__global__ void softmax_reduce_kernel(const float* __restrict__ logits,
                                      float* __restrict__ red)
{
  __shared__ float sm[1024];
  const int t = threadIdx.x;
  float m = -INFINITY;
  for (int v = t; v < VOCAB; v += 1024) m = fmaxf(m, logits[v]);
  sm[t] = m;
  __syncthreads();
  for (int s = 512; s > 0; s >>= 1) {
    if (t < s) sm[t] = fmaxf(sm[t], sm[t + s]);
    __syncthreads();
  }
  const float M = sm[0];
  __syncthreads();
  float sum = 0.0f;
  for (int v = t; v < VOCAB; v += 1024) sum += expf(logits[v] - M);
  sm[t] = sum;
  __syncthreads();
  for (int s = 512; s > 0; s >>= 1) {
    if (t < s) sm[t] += sm[t + s];
    __syncthreads();
  }
  if (t == 0) { red[0] = M; red[1] = logf(sm[0]); }
}

// ---- 7) logprobs -> d_out ---------------------------------------------------
__global__ void logprob_kernel(const float* __restrict__ logits,
                               const float* __restrict__ red,
                               float* __restrict__ out)
{
  const int v = blockIdx.x * blockDim.x + threadIdx.x;
  if (v < VOCAB) out[v] = logits[v] - red[0] - red[1];
}

extern "C" void kernel_launch(void* const* d_in, const int* in_sizes, int n_in,
                              void* d_out, int out_size, void* d_ws, size_t ws_size,
                              hipStream_t stream) {
  const int*   token = (const int*)  d_in[0];
  const float* h_in  = (const float*)d_in[1];
  const float* c_in  = (const float*)d_in[2];
  const float* emb   = (const float*)d_in[3];
  const float* Wih   = (const float*)d_in[4];
  const float* Whh   = (const float*)d_in[5];
  const float* bih   = (const float*)d_in[6];
  const float* bhh   = (const float*)d_in[7];
  const float* Wout  = (const float*)d_in[8];
  const float* bout  = (const float*)d_in[9];
  float* out = (float*)d_out;          // [logprobs(50257) | h_new(2048) | c_new(2048)]
  float* ws  = (float*)d_ws;

  // ws layout (floats), all region bases 8B-aligned:
  float* x      = ws;                                   // 2048
  float* hcopy  = x + HIDDEN;                           // 2048
  float* gpart  = hcopy + HIDDEN;                       // 8 * 8192  = 65536
  float* lpart  = gpart + (size_t)KSPLIT_G * GATES;     // 4 * 50264 = 201056
  float* logits = lpart + (size_t)KSPLIT_V * VPAD;      // 50264
  float* red    = logits + VPAD;                        // 2
  // total ~320954 floats (~1.23 MB) of d_ws

  // 1) embedding gather + relu
  embed_relu_kernel<<<HIDDEN / 256, 256, 0, stream>>>(token, emb, x);

  // 2) gates GEMV: 256 row-tiles x 8 K-segs = 2048 waves, 4 waves/block
  gates_gemv_kernel<<<(GATES / 32) * KSPLIT_G / 4, 128, 0, stream>>>(
      Wih, Whh, x, h_in, gpart);

  // 3) LSTM pointwise -> h_new/c_new into d_out (+ aligned h copy in ws)
  lstm_cell_kernel<<<HIDDEN / 256, 256, 0, stream>>>(
      gpart, bih, bhh, c_in, out, hcopy);

  // 4) logits GEMV: 1571 row-tiles x 4 K-segs = 6284 waves, 4 waves/block
  {
    const int nt = (VOCAB + 31) / 32;            // 1571
    const int blocks = nt;                       // 4 waves/block = 4 K-segs
    logits_gemv_kernel<<<blocks * KSPLIT_V / 4, 128, 0, stream>>>(
        Wout, hcopy, lpart);
  }

  // 5) combine partials + bias
  logits_combine_kernel<<<(VOCAB + 255) / 256, 256, 0, stream>>>(
      lpart, bout, logits);

  // 6) max + log-sum-exp
  softmax_reduce_kernel<<<1, 1024, 0, stream>>>(logits, red);

  // 7) log-probs to d_out
  logprob_kernel<<<(VOCAB + 255) / 256, 256, 0, stream>>>(logits, red, out);
}